// MGSKDLoss_7765300871713
// MI455X (gfx1250) — compile-verified
//
#include <hip/hip_runtime.h>
#include <math.h>

// Problem constants from the reference (setup_inputs: L=4, B=16, S=256, D=768)
#define BDIM   16
#define SDIM   256
#define DDIM   768
#define HREL   64
#define DH     12          // DDIM / HREL
#define KK     20          // K1 == K2 == 20 (S=256 so min() is 20)
#define LAYERS 4
#define MTOK   2

typedef __attribute__((ext_vector_type(2))) float v2f;
typedef __attribute__((ext_vector_type(8))) float v8f;
typedef int v4i_vs __attribute__((vector_size(16)));   // matches builtin's V4i pointee

// Exact f32 WMMA: D = A(16x4) * B(4x16) + C. K=12 -> 3 chained calls.
__device__ __forceinline__ v8f wmma_f32_k4(v2f a, v2f b, v8f c) {
  return __builtin_amdgcn_wmma_f32_16x16x4_f32(
      /*neg_a=*/false, a, /*neg_b=*/false, b,
      /*c_mod=*/(short)0, c, /*reuse_a=*/false, /*reuse_b=*/false);
}

// ---- CDNA5 async global->LDS copy path (CDNA5 ISA) -----------
#if defined(__has_builtin)
#if __has_builtin(__builtin_amdgcn_global_load_async_to_lds_b128)
#define HAVE_ASYNC_LDS 1
#endif
#endif

__device__ __forceinline__ void async_wait0() {
#if defined(__has_builtin) && __has_builtin(__builtin_amdgcn_s_wait_asynccnt)
  __builtin_amdgcn_s_wait_asynccnt(0);
#else
  asm volatile("s_wait_asynccnt 0x0" ::: "memory");
#endif
}

#ifdef HAVE_ASYNC_LDS
__device__ __forceinline__ void async_copy16(const float* gptr, float* lptr) {
  __builtin_amdgcn_global_load_async_to_lds_b128(
      (__attribute__((address_space(1))) v4i_vs*)(const_cast<float*>(gptr)),
      (__attribute__((address_space(3))) v4i_vs*)(lptr),
      /*offset=*/0, /*cpol=*/0);
}
#endif

__device__ __forceinline__ float smooth_l1(float x, float y) {
  float d = x - y, ad = fabsf(d);
  return (ad < 1.f) ? 0.5f * d * d : ad - 0.5f;
}

__global__ void zero_kernel(float* p, int n) {
  int i = blockIdx.x * blockDim.x + threadIdx.x;
  if (i < n) p[i] = 0.f;
}

// ---------------------------------------------------------------------------
// Fused pair-interaction kernel. Block = one (b, 16-row i-strip), 256 thr (8 waves).
// Loops h=0..63: stages 256x12 head panels of s and t in LDS via
// GLOBAL_LOAD_ASYNC_TO_LDS_B128, computes 16x256 score strips for BOTH tensors
// via V_WMMA_F32_16X16X4_F32, accumulates sum((s_pair - t_pair)^2) into acc[0],
// and the t-softmax (masked) into the per-strip `local` accumulator (summed
// over heads). Never spills the SxS score tensors (would be 268 MB/tensor).
// ---------------------------------------------------------------------------
__global__ __launch_bounds__(256) void pair_kernel(
    const float* __restrict__ s, const float* __restrict__ t,
    const float* __restrict__ mask, float* __restrict__ local,
    float* __restrict__ acc)
{
  __shared__ __align__(16) float sp[SDIM * DH];   // s head panel (256 x 12)
  __shared__ __align__(16) float tp[SDIM * DH];   // t head panel
  __shared__ float tstrip[16][SDIM];              // t softmax-input strip
  __shared__ float lstrip[16][SDIM];              // local accumulator strip
  __shared__ float rowred[16][16];
  __shared__ float rowmax[16];
  __shared__ float rowsum[16];
  __shared__ float blocksum;

  const int strip  = blockIdx.x;
  const int b      = blockIdx.y;
  const int i_base = strip * 16;
  const int tid    = threadIdx.x;
  const int wave   = tid >> 5;           // wave32: 8 waves / block
  const int lane   = tid & 31;
  const int half   = lane >> 4;          // lanes 16-31 carry K+2 / M+8 halves
  const int mrow   = lane & 15;
  const float rsd  = 0.28867513459481287f; // 1/sqrt(12)

  for (int idx = tid; idx < 16 * SDIM; idx += 256)
    lstrip[idx >> 8][idx & 255] = 0.f;
  if (tid == 0) blocksum = 0.f;

  float sqacc = 0.f;

  for (int h = 0; h < HREL; ++h) {
    __syncthreads();  // previous iteration consumers done
#ifdef HAVE_ASYNC_LDS
    // 256x12 f32 panel == 768 16-byte segments per tensor; LDS-direct copy.
    for (int q = tid; q < SDIM * 3; q += 256) {
      int row = q / 3, seg = q - row * 3;
      size_t go = ((size_t)b * SDIM + row) * DDIM + h * DH + seg * 4;
      async_copy16(&s[go], &sp[row * DH + seg * 4]);
      async_copy16(&t[go], &tp[row * DH + seg * 4]);
    }
#else
    for (int idx = tid; idx < SDIM * DH; idx += 256) {
      int row = idx / DH, k = idx - row * DH;
      size_t go = ((size_t)b * SDIM + row) * DDIM + h * DH + k;
      sp[idx] = s[go];
      tp[idx] = t[go];
    }
#endif
    if (h + 1 < HREL) {   // global_prefetch_b8 of next head panel
      size_t go = ((size_t)b * SDIM + tid) * DDIM + (h + 1) * DH;
      __builtin_prefetch(&s[go], 0, 1);
      __builtin_prefetch(&t[go], 0, 1);
    }
#ifdef HAVE_ASYNC_LDS
    async_wait0();        // ASYNCcnt==0: LDS writes landed before barrier
#endif
    __syncthreads();

    // Each wave owns 2 of the 16 j-tiles; uniform control flow (EXEC all 1s).
    #pragma unroll
    for (int u = 0; u < 2; ++u) {
      const int jt = wave + u * 8;
      const int j_base = jt * 16;
      v8f sc = {};
      v8f tc = {};
      #pragma unroll
      for (int kc = 0; kc < 3; ++kc) {
        // ISA A/B layout: lanes 0-15 -> K=kb,kb+1 ; lanes 16-31 -> K=kb+2,kb+3
        const int ka0 = kc * 4 + (half ? 2 : 0);
        const int ka1 = ka0 + 1;
        v2f a, bb, at, bt;
        a.x  = sp[(i_base + mrow) * DH + ka0];
        a.y  = sp[(i_base + mrow) * DH + ka1];
        bb.x = sp[(j_base + mrow) * DH + ka0];
        bb.y = sp[(j_base + mrow) * DH + ka1];
        sc = wmma_f32_k4(a, bb, sc);
        at.x = tp[(i_base + mrow) * DH + ka0];
        at.y = tp[(i_base + mrow) * DH + ka1];
        bt.x = tp[(j_base + mrow) * DH + ka0];
        bt.y = tp[(j_base + mrow) * DH + ka1];
        tc = wmma_f32_k4(at, bt, tc);
      }
      const int j  = j_base + mrow;      // D layout: N = lane&15
      const float mj = mask[b * SDIM + j];
      #pragma unroll
      for (int v = 0; v < 8; ++v) {
        const int ir = v + 8 * half;     // D layout: M = v + 8*half
        const int i  = i_base + ir;
        const float m  = mask[b * SDIM + i] * mj;
        const float ss = sc[v] * rsd * m;     // s_pair (masked score)
        const float ttv = tc[v] * rsd * m;    // t_pair
        const float d = ss - ttv;
        sqacc += d * d;
        tstrip[ir][j] = ttv + (1.f - m) * (-10000.f); // softmax input
      }
    }
    __syncthreads();

    { // row max (16 rows x 16 threads each)
      int r = tid >> 4, c = tid & 15;
      float mx = -3.4e38f;
      for (int j = c; j < SDIM; j += 16) mx = fmaxf(mx, tstrip[r][j]);
      rowred[r][c] = mx;
    }
    __syncthreads();
    if (tid < 16) {
      float mx = rowred[tid][0];
      for (int c = 1; c < 16; ++c) mx = fmaxf(mx, rowred[tid][c]);
      rowmax[tid] = mx;
    }
    __syncthreads();
    { // row sum of exp
      int r = tid >> 4, c = tid & 15;
      float sum = 0.f, mx = rowmax[r];
      for (int j = c; j < SDIM; j += 16) sum += __expf(tstrip[r][j] - mx);
      rowred[r][c] = sum;
    }
    __syncthreads();
    if (tid < 16) {
      float sum = 0.f;
      for (int c = 0; c < 16; ++c) sum += rowred[tid][c];
      rowsum[tid] = sum;
    }
    __syncthreads();
    for (int idx = tid; idx < 16 * SDIM; idx += 256) {
      int r = idx >> 8, j = idx & 255;
      float m  = mask[b * SDIM + i_base + r] * mask[b * SDIM + j];
      float sm = __expf(tstrip[r][j] - rowmax[r]) / rowsum[r] * m;
      lstrip[r][j] += sm;   // local = sum over heads of masked softmax
    }
  }
  __syncthreads();

  for (int idx = tid; idx < 16 * SDIM; idx += 256) {
    int r = idx >> 8, j = idx & 255;
    local[((size_t)b * SDIM + i_base + r) * SDIM + j] = lstrip[r][j];
  }
  atomicAdd(&blocksum, sqacc);
  __syncthreads();
  if (tid == 0) atomicAdd(&acc[0], blocksum);
}

// g[b,j] = sum_i local[b,i,j]  (== sum over h,i of sm)
__global__ void colsum_kernel(const float* __restrict__ local, float* __restrict__ g) {
  int b = blockIdx.x, j = threadIdx.x;
  float s = 0.f;
  for (int i = 0; i < SDIM; ++i) s += local[((size_t)b * SDIM + i) * SDIM + j];
  g[b * SDIM + j] = s;
}

// Stable top-20 of g rows (ties -> lowest index, matching jax.lax.top_k)
__global__ void topk_g_kernel(const float* __restrict__ g, int* __restrict__ gidx) {
  int b = threadIdx.x;
  if (b >= BDIM) return;
  unsigned excl[SDIM / 32] = {};
  const float* row = g + b * SDIM;
  for (int r = 0; r < KK; ++r) {
    float best = -3.4e38f; int bj = 0;
    for (int j = 0; j < SDIM; ++j) {
      if (excl[j >> 5] & (1u << (j & 31))) continue;
      float v = row[j];
      if (v > best) { best = v; bj = j; }
    }
    excl[bj >> 5] |= 1u << (bj & 31);
    gidx[b * KK + r] = bj;
  }
}

// top-20 of local rows selected by gidx, with diagonal zeroed (1 - eye)
__global__ void topk_local_kernel(const float* __restrict__ local,
                                  const int* __restrict__ gidx,
                                  int* __restrict__ lidx) {
  int b = blockIdx.x, t = threadIdx.x;
  if (t >= KK) return;
  int r = gidx[b * KK + t];
  const float* row = local + ((size_t)b * SDIM + r) * SDIM;
  unsigned excl[SDIM / 32] = {};
  for (int k = 0; k < KK; ++k) {
    float best = -3.4e38f; int bj = 0;
    for (int j = 0; j < SDIM; ++j) {
      if (excl[j >> 5] & (1u << (j & 31))) continue;
      float v = (j == r) ? 0.f : row[j];
      if (v > best) { best = v; bj = j; }
    }
    excl[bj >> 5] |= 1u << (bj & 31);
    lidx[(b * KK + t) * KK + k] = bj;
  }
}

// Triplet angle: per (b, gi) block computes 20 normalized 768-d diff vectors
// for s and t, their 20x20 Grams, masks, and smooth-L1 accumulation.
__global__ __launch_bounds__(256) void triplet_kernel(
    const float* __restrict__ s, const float* __restrict__ t,
    const float* __restrict__ mask, const int* __restrict__ gidx,
    const int* __restrict__ lidx, float* __restrict__ acc, int slot)
{
  __shared__ float nvec[KK * DDIM];     // 61440 B, reused for s then t
  __shared__ float angS[KK * KK];
  __shared__ float angT[KK * KK];
  __shared__ float norm2[KK];
  __shared__ float ninv[KK];
  __shared__ int   larr[KK];
  __shared__ float meArr[KK];
  __shared__ float rnum, rden;

  const int gi = blockIdx.x, b = blockIdx.y, tid = threadIdx.x;
  const int g = gidx[b * KK + gi];
  if (tid < KK) {
    int li = lidx[(b * KK + gi) * KK + tid];
    larr[tid]  = li;
    meArr[tid] = mask[b * SDIM + g] + mask[b * SDIM + li];
  }
  if (tid == 0) { rnum = 0.f; rden = 0.f; }
  __syncthreads();

  for (int tt = 0; tt < 2; ++tt) {
    const float* rep = tt ? t : s;
    float* ang = tt ? angT : angS;
    if (tid < KK) norm2[tid] = 0.f;
    __syncthreads();
    for (int idx = tid; idx < KK * DDIM; idx += 256) {
      int m = idx / DDIM, d = idx - m * DDIM;
      float diff = rep[((size_t)b * SDIM + g) * DDIM + d]
                 - rep[((size_t)b * SDIM + larr[m]) * DDIM + d];
      nvec[idx] = diff;
      atomicAdd(&norm2[m], diff * diff);
    }
    __syncthreads();
    if (tid < KK) ninv[tid] = 1.f / fmaxf(sqrtf(norm2[tid]), 1e-12f);
    __syncthreads();
    for (int idx = tid; idx < KK * DDIM; idx += 256)
      nvec[idx] *= ninv[idx / DDIM];
    __syncthreads();
    for (int p = tid; p < KK * KK; p += 256) {
      int m = p / KK, n = p - m * KK;
      const float* a = &nvec[m * DDIM];
      const float* c = &nvec[n * DDIM];
      float dot = 0.f;
      for (int d = 0; d < DDIM; ++d) dot += a[d] * c[d];
      ang[p] = dot;
    }
    __syncthreads();
  }

  float num = 0.f, den = 0.f;
  for (int p = tid; p < KK * KK; p += 256) {
    int m = p / KK, n = p - m * KK;
    float amb = (meArr[m] * meArr[n] == 4.f) ? 1.f : 0.f;
    float eye = (m == n) ? 0.f : 1.f;
    float as_ = angS[p], at_ = angT[p];
    float ams = amb * eye * ((as_ != 0.f) ? 1.f : 0.f);
    float amt = amb * eye * ((at_ != 0.f) ? 1.f : 0.f);
    num += smooth_l1(as_ * ams, at_ * amt);
    den += ams;
  }
  atomicAdd(&rnum, num);
  atomicAdd(&rden, den);
  __syncthreads();
  if (tid == 0) {
    atomicAdd(&acc[1 + 2 * slot], rnum);
    atomicAdd(&acc[2 + 2 * slot], rden);
  }
}

// Mean-pool s and t over the sequence with the mask.
__global__ void pool_kernel(const float* __restrict__ s, const float* __restrict__ t,
                            const float* __restrict__ mask,
                            float* __restrict__ ps, float* __restrict__ pt)
{
  int b = blockIdx.x;
  float msum = 0.f;
  for (int i = 0; i < SDIM; ++i) msum += mask[b * SDIM + i];
  float inv = 1.f / msum;
  for (int d = threadIdx.x; d < DDIM; d += blockDim.x) {
    float as = 0.f, at = 0.f;
    for (int i = 0; i < SDIM; ++i) {
      float m = mask[b * SDIM + i];
      as += s[((size_t)b * SDIM + i) * DDIM + d] * m;
      at += t[((size_t)b * SDIM + i) * DDIM + d] * m;
    }
    ps[b * DDIM + d] = as * inv;
    pt[b * DDIM + d] = at * inv;
  }
}

// Sample-loss angles: per head h, 16x16 normalized 12-d diffs for s and t,
// then sum of smooth_l1 over the (16,16,16) angle tensor.
__global__ __launch_bounds__(256) void sample_angle_kernel(
    const float* __restrict__ ps, const float* __restrict__ pt,
    float* __restrict__ acc)
{
  __shared__ float vs[BDIM * DH];
  __shared__ float vt[BDIM * DH];
  __shared__ float ns[BDIM * BDIM * DH];
  __shared__ float nt[BDIM * BDIM * DH];
  __shared__ float rsum;
  const int h = blockIdx.x, tid = threadIdx.x;
  if (tid == 0) rsum = 0.f;
  for (int idx = tid; idx < BDIM * DH; idx += 256) {
    int bb = idx / DH, d = idx - bb * DH;
    vs[idx] = ps[bb * DDIM + h * DH + d];
    vt[idx] = pt[bb * DDIM + h * DH + d];
  }
  __syncthreads();
  { // one thread per (i,j) pair: 256 threads exactly
    int i = tid >> 4, j = tid & 15;
    float ds_[DH], dt_[DH], n2s = 0.f, n2t = 0.f;
    #pragma unroll
    for (int d = 0; d < DH; ++d) {
      float a = vs[i * DH + d] - vs[j * DH + d];
      float c = vt[i * DH + d] - vt[j * DH + d];
      ds_[d] = a; dt_[d] = c; n2s += a * a; n2t += c * c;
    }
    float is_ = 1.f / fmaxf(sqrtf(n2s), 1e-12f);
    float it_ = 1.f / fmaxf(sqrtf(n2t), 1e-12f);
    #pragma unroll
    for (int d = 0; d < DH; ++d) {
      ns[tid * DH + d] = ds_[d] * is_;
      nt[tid * DH + d] = dt_[d] * it_;
    }
  }
  __syncthreads();
  float lsum = 0.f;
  for (int idx = tid; idx < BDIM * BDIM * BDIM; idx += 256) {
    int i = idx >> 8, j = (idx >> 4) & 15, k = idx & 15;
    const float* a = &ns[(i * 16 + j) * DH];
    const float* c = &ns[(i * 16 + k) * DH];
    const float* e = &nt[(i * 16 + j) * DH];
    const float* f = &nt[(i * 16 + k) * DH];
    float sa = 0.f, ta = 0.f;
    #pragma unroll
    for (int d = 0; d < DH; ++d) { sa += a[d] * c[d]; ta += e[d] * f[d]; }
    lsum += smooth_l1(sa, ta);
  }
  atomicAdd(&rsum, lsum);
  __syncthreads();
  if (tid == 0) atomicAdd(&acc[5], rsum);
}

__global__ void final_kernel(const float* __restrict__ mask,
                             const float* __restrict__ acc,
                             float* __restrict__ out)
{
  __shared__ float part[BDIM];
  int tid = threadIdx.x;
  if (tid < BDIM) {
    float s = 0.f;
    for (int i = 0; i < SDIM; ++i) s += mask[tid * SDIM + i];
    part[tid] = s * s;
  }
  __syncthreads();
  if (tid == 0) {
    float msq = 0.f;
    for (int b = 0; b < BDIM; ++b) msq += part[b];
    float pair_den = (float)HREL * msq;                 // sum(mask_ext), same both layers
    float token = acc[0] / pair_den
                + acc[1] / acc[2]
                + acc[3] / acc[4];
    float sample = acc[5] / (float)(HREL * BDIM * BDIM * BDIM);
    out[0] = 4.0f * sample + 1.0f * token;              // W1*sample + W2*token
  }
}

extern "C" void kernel_launch(void* const* d_in, const int* in_sizes, int n_in,
                              void* d_out, int out_size, void* d_ws, size_t ws_size,
                              hipStream_t stream)
{
  (void)in_sizes; (void)n_in; (void)out_size; (void)ws_size;
  const float* S    = (const float*)d_in[0];
  const float* T    = (const float*)d_in[1];
  const float* mask = (const float*)d_in[2];
  float* out = (float*)d_out;

  // Workspace layout (~4.34 MB total)
  float* ws    = (float*)d_ws;
  float* acc   = ws;                                     // 16 floats
  float* local = ws + 16;                                // B*S*S
  float* g     = local + (size_t)BDIM * SDIM * SDIM;     // B*S
  int*   gidx  = (int*)(g + BDIM * SDIM);                // B*KK
  int*   lidx  = gidx + BDIM * KK;                       // B*KK*KK
  float* ps    = (float*)(lidx + BDIM * KK * KK);        // B*D
  float* pt    = ps + BDIM * DDIM;                       // B*D

  const size_t layer_stride = (size_t)BDIM * SDIM * DDIM;

  zero_kernel<<<1, 64, 0, stream>>>(acc, 16);

  for (int l = 0; l < MTOK; ++l) {                       // token-phrase layers
    const float* sl = S + l * layer_stride;
    const float* tl = T + l * layer_stride;
    const int nz = BDIM * SDIM * SDIM + BDIM * SDIM;     // local + g contiguous
    zero_kernel<<<(nz + 255) / 256, 256, 0, stream>>>(local, nz);
    pair_kernel<<<dim3(SDIM / 16, BDIM), 256, 0, stream>>>(sl, tl, mask, local, acc);
    colsum_kernel<<<BDIM, SDIM, 0, stream>>>(local, g);
    topk_g_kernel<<<1, 32, 0, stream>>>(g, gidx);
    topk_local_kernel<<<BDIM, 32, 0, stream>>>(local, gidx, lidx);
    triplet_kernel<<<dim3(KK, BDIM), 256, 0, stream>>>(sl, tl, mask, gidx, lidx, acc, l);
  }
  for (int l = MTOK; l < LAYERS; ++l) {                  // sample layers
    const float* sl = S + l * layer_stride;
    const float* tl = T + l * layer_stride;
    pool_kernel<<<BDIM, 256, 0, stream>>>(sl, tl, mask, ps, pt);
    sample_angle_kernel<<<HREL, 256, 0, stream>>>(ps, pt, acc);
  }
  final_kernel<<<1, 32, 0, stream>>>(mask, acc, out);
}